// NLQR_73521250173166
// MI455X (gfx1250) — compile-verified
//
#include <hip/hip_runtime.h>

// LQR backward/forward solve for MI455X (gfx1250, wave32).
// One wave = one batch. Backward: Qt = Q0 + Fp^T * Vp * Fp done with chained
// V_WMMA_F32_16X16X4_F32 (fp32 WMMA, 3 K-slices per product since Fp rows
// 12..15 are zero). Forward: lane-parallel matvecs + quadratic cost.
// Memory-bound problem (~480MB streamed @ 23.3 TB/s => ~21us floor).

#define kNB 1024
#define kT  128
#define kNS 12
#define kNC 4
#define kN  16
#define WPB 8   // waves per block (256 threads)

typedef float v2f __attribute__((ext_vector_type(2)));
typedef float v8f __attribute__((ext_vector_type(8)));

__device__ __forceinline__ void lds_fence() {
  // DS ops from one wave execute in order; this stops the compiler from
  // reordering cross-lane LDS traffic and waits for outstanding DS ops.
  asm volatile("s_wait_dscnt 0" ::: "memory");
}

// Adjugate-based 4x4 inverse (Quu is symmetric positive definite here,
// so pinv == inv). All lanes compute it redundantly.
__device__ __forceinline__ void inv4x4(const float* m, float* inv) {
  inv[0]  =  m[5]*m[10]*m[15] - m[5]*m[11]*m[14] - m[9]*m[6]*m[15]
           + m[9]*m[7]*m[14] + m[13]*m[6]*m[11] - m[13]*m[7]*m[10];
  inv[4]  = -m[4]*m[10]*m[15] + m[4]*m[11]*m[14] + m[8]*m[6]*m[15]
           - m[8]*m[7]*m[14] - m[12]*m[6]*m[11] + m[12]*m[7]*m[10];
  inv[8]  =  m[4]*m[9]*m[15] - m[4]*m[11]*m[13] - m[8]*m[5]*m[15]
           + m[8]*m[7]*m[13] + m[12]*m[5]*m[11] - m[12]*m[7]*m[9];
  inv[12] = -m[4]*m[9]*m[14] + m[4]*m[10]*m[13] + m[8]*m[5]*m[14]
           - m[8]*m[6]*m[13] - m[12]*m[5]*m[10] + m[12]*m[6]*m[9];
  inv[1]  = -m[1]*m[10]*m[15] + m[1]*m[11]*m[14] + m[9]*m[2]*m[15]
           - m[9]*m[3]*m[14] - m[13]*m[2]*m[11] + m[13]*m[3]*m[10];
  inv[5]  =  m[0]*m[10]*m[15] - m[0]*m[11]*m[14] - m[8]*m[2]*m[15]
           + m[8]*m[3]*m[14] + m[12]*m[2]*m[11] - m[12]*m[3]*m[10];
  inv[9]  = -m[0]*m[9]*m[15] + m[0]*m[11]*m[13] + m[8]*m[1]*m[15]
           - m[8]*m[3]*m[13] - m[12]*m[1]*m[11] + m[12]*m[3]*m[9];
  inv[13] =  m[0]*m[9]*m[14] - m[0]*m[10]*m[13] - m[8]*m[1]*m[14]
           + m[8]*m[2]*m[13] + m[12]*m[1]*m[10] - m[12]*m[2]*m[9];
  inv[2]  =  m[1]*m[6]*m[15] - m[1]*m[7]*m[14] - m[5]*m[2]*m[15]
           + m[5]*m[3]*m[14] + m[13]*m[2]*m[7] - m[13]*m[3]*m[6];
  inv[6]  = -m[0]*m[6]*m[15] + m[0]*m[7]*m[14] + m[4]*m[2]*m[15]
           - m[4]*m[3]*m[14] - m[12]*m[2]*m[7] + m[12]*m[3]*m[6];
  inv[10] =  m[0]*m[5]*m[15] - m[0]*m[7]*m[13] - m[4]*m[1]*m[15]
           + m[4]*m[3]*m[13] + m[12]*m[1]*m[7] - m[12]*m[3]*m[5];
  inv[14] = -m[0]*m[5]*m[14] + m[0]*m[6]*m[13] + m[4]*m[1]*m[14]
           - m[4]*m[2]*m[13] - m[12]*m[1]*m[6] + m[12]*m[2]*m[5];
  inv[3]  = -m[1]*m[6]*m[11] + m[1]*m[7]*m[10] + m[5]*m[2]*m[11]
           - m[5]*m[3]*m[10] - m[9]*m[2]*m[7] + m[9]*m[3]*m[6];
  inv[7]  =  m[0]*m[6]*m[11] - m[0]*m[7]*m[10] - m[4]*m[2]*m[11]
           + m[4]*m[3]*m[10] + m[8]*m[2]*m[7] - m[8]*m[3]*m[6];
  inv[11] = -m[0]*m[5]*m[11] + m[0]*m[7]*m[9] + m[4]*m[1]*m[11]
           - m[4]*m[3]*m[9] - m[8]*m[1]*m[7] + m[8]*m[3]*m[5];
  inv[15] =  m[0]*m[5]*m[10] - m[0]*m[6]*m[9] - m[4]*m[1]*m[10]
           + m[4]*m[2]*m[9] + m[8]*m[1]*m[6] - m[8]*m[2]*m[5];
  float det = m[0]*inv[0] + m[1]*inv[4] + m[2]*inv[8] + m[3]*inv[12];
  float invdet = 1.0f / det;
  #pragma unroll
  for (int i = 0; i < 16; ++i) inv[i] *= invdet;
}

// ---------------------------------------------------------------------------
// Backward Riccati scan: per wave (=batch), t = T-1 .. 0.
//   Fp = [[A | B]; 0]  (16x16, rows 12..15 zero)
//   T1 = Vp * Fp                 (3x v_wmma_f32_16x16x4_f32)
//   Qt = Q0 + Fp^T * T1          (3x v_wmma_f32_16x16x4_f32, C preloaded Q0)
//   Quu^-1 (4x4 adjugate), k_t = -Quu^-1 qu  -> workspace
//   V' = Qxx - Qxu Quu^-1 Qux,  v' = qx + Qxu k_t   (lane-per-column)
// ---------------------------------------------------------------------------
__global__ __launch_bounds__(256) void lqr_backward(
    const float* __restrict__ Q, const float* __restrict__ p,
    const float* __restrict__ A, const float* __restrict__ Bm,
    float* __restrict__ kout)
{
  __shared__ float Vbuf[WPB][kN * kN];
  __shared__ float Tbuf[WPB][kN * kN];
  __shared__ float Qbuf[WPB][kN * kN];
  __shared__ float vbuf[WPB][kN];
  __shared__ float qbuf[WPB][kN];

  const int wid  = threadIdx.x >> 5;
  const int lane = threadIdx.x & 31;
  const int b    = blockIdx.x * WPB + wid;
  const int n    = lane & 15;   // column / M index within the 16x16 tile
  const int h    = lane >> 4;   // wave half (selects K sub-pairs)

  float* V  = Vbuf[wid];
  float* Tm = Tbuf[wid];
  float* Qs = Qbuf[wid];
  float* vv = vbuf[wid];
  float* qv = qbuf[wid];

  // V_T = 0, v_T = 0 (padding rows/cols 12..15 stay zero forever)
  #pragma unroll
  for (int i = 0; i < 8; ++i) V[i * 32 + lane] = 0.0f;
  if (lane < kN) vv[lane] = 0.0f;
  lds_fence();

  for (int t = kT - 1; t >= 0; --t) {
    const size_t bt = (size_t)b * kT + t;
    const float* __restrict__ Qb = Q  + bt * (kN * kN);
    const float* __restrict__ pb = p  + bt * kN;
    const float* __restrict__ Ab = A  + bt * (kNS * kNS);
    const float* __restrict__ Bb = Bm + bt * (kNS * kNC);

    // Prefetch next timestep's Q and A blocks into the cache hierarchy.
    if (t > 0) {
      __builtin_prefetch((const char*)(Qb - kN * kN) + lane * 64, 0, 1);
      __builtin_prefetch((const char*)(Ab - kNS * kNS) + lane * 32, 0, 1);
    }

    // Per-lane column view of F: F[r][n] = fcol[r * fstride]
    const float* __restrict__ fcol = (n < kNS) ? (Ab + n) : (Bb + (n - kNS));
    const int fstride = (n < kNS) ? kNS : kNC;

    // F slices. A-layout of Fp^T == B-layout of Fp (both hold
    // Fp[4s+2h+c][n] in lane), so these serve both WMMA chains.
    // Slice 3 (K=12..15) is all-zero -> skipped entirely.
    v2f fsl[3];
    #pragma unroll
    for (int s = 0; s < 3; ++s) {
      const int k0 = 4 * s + 2 * h;
      fsl[s].x = fcol[k0 * fstride];
      fsl[s].y = fcol[(k0 + 1) * fstride];
    }

    // ---- T1 = Vp * Fp
    v8f acc1;
    #pragma unroll
    for (int i = 0; i < 8; ++i) acc1[i] = 0.0f;
    #pragma unroll
    for (int s = 0; s < 3; ++s) {
      const int k0 = 4 * s + 2 * h;
      v2f a;                      // A-layout of Vp: Vp[n][k0], Vp[n][k0+1]
      a.x = V[n * kN + k0];
      a.y = V[n * kN + k0 + 1];
      acc1 = __builtin_amdgcn_wmma_f32_16x16x4_f32(
          false, a, false, fsl[s], (short)0, acc1, false, false);
    }
    // scatter T1 (C-layout) into LDS so it can be re-read in B-layout
    #pragma unroll
    for (int v = 0; v < 8; ++v) Tm[(v + 8 * h) * kN + n] = acc1[v];
    lds_fence();

    // ---- Qt = Q0 + Fp^T * T1  (Q0 preloaded into the accumulator)
    v8f acc;
    #pragma unroll
    for (int v = 0; v < 8; ++v) acc[v] = Qb[(v + 8 * h) * kN + n];
    #pragma unroll
    for (int s = 0; s < 3; ++s) {
      const int k0 = 4 * s + 2 * h;
      v2f bv;                     // B-layout of T1: rows k0, k0+1
      bv.x = Tm[k0 * kN + n];
      bv.y = Tm[(k0 + 1) * kN + n];
      acc = __builtin_amdgcn_wmma_f32_16x16x4_f32(
          false, fsl[s], false, bv, (short)0, acc, false, false);
    }
    #pragma unroll
    for (int v = 0; v < 8; ++v) Qs[(v + 8 * h) * kN + n] = acc[v];

    // qt[n] = p[n] + sum_s F[s][n] * v[s]   (lanes 0..15)
    if (h == 0) {
      float q = pb[n];
      #pragma unroll
      for (int s = 0; s < kNS; ++s) q += fcol[s * fstride] * vv[s];
      qv[n] = q;
    }
    lds_fence();

    // ---- scalar phase: Quu^-1, k, V', v'
    float M[16], Wi[16];
    #pragma unroll
    for (int i = 0; i < 4; ++i)
      #pragma unroll
      for (int j = 0; j < 4; ++j)
        M[i * 4 + j] = Qs[(kNS + i) * kN + (kNS + j)];
    inv4x4(M, Wi);

    float qu[4];
    #pragma unroll
    for (int j = 0; j < 4; ++j) qu[j] = qv[kNS + j];
    float kt[4];
    #pragma unroll
    for (int u = 0; u < 4; ++u)
      kt[u] = -(Wi[u * 4 + 0] * qu[0] + Wi[u * 4 + 1] * qu[1] +
                Wi[u * 4 + 2] * qu[2] + Wi[u * 4 + 3] * qu[3]);
    if (lane < kNC) kout[bt * kNC + lane] = kt[lane];

    if (lane < kNS) {
      const int s = lane;
      float qux[4];               // Qux[:,s]
      #pragma unroll
      for (int j = 0; j < 4; ++j) qux[j] = Qs[(kNS + j) * kN + s];
      float g[4];                 // g = Quu^-1 * Qux[:,s]
      #pragma unroll
      for (int u = 0; u < 4; ++u)
        g[u] = Wi[u * 4 + 0] * qux[0] + Wi[u * 4 + 1] * qux[1] +
               Wi[u * 4 + 2] * qux[2] + Wi[u * 4 + 3] * qux[3];
      #pragma unroll
      for (int i = 0; i < kNS; ++i) {
        float val = Qs[i * kN + s];           // Qxx[i][s]
        #pragma unroll
        for (int u = 0; u < 4; ++u)
          val -= Qs[i * kN + kNS + u] * g[u]; // - Qxu[i][:] . g
        V[i * kN + s] = val;
      }
      float vn = qv[s];
      #pragma unroll
      for (int u = 0; u < 4; ++u) vn += Qs[s * kN + kNS + u] * kt[u];
      vv[s] = vn;
    }
    lds_fence();
  }
}

// ---------------------------------------------------------------------------
// Forward rollout: per wave (=batch), t = 0 .. T-1.
// xu held across lanes 0..15 (lanes>=16 carry zero); matvecs via __shfl
// broadcasts; cost accumulated per-lane, reduced once at the end.
// ---------------------------------------------------------------------------
__global__ __launch_bounds__(256) void lqr_forward(
    const float* __restrict__ x_init, const float* __restrict__ cu,
    const float* __restrict__ Q, const float* __restrict__ p,
    const float* __restrict__ A, const float* __restrict__ Bm,
    const float* __restrict__ kin,
    float* __restrict__ xs, float* __restrict__ us, float* __restrict__ cost)
{
  const int wid  = threadIdx.x >> 5;
  const int lane = threadIdx.x & 31;
  const int b    = blockIdx.x * WPB + wid;
  const int ln   = lane & 15;

  float x = (lane < kNS) ? x_init[(size_t)b * kNS + lane] : 0.0f;
  float cacc = 0.0f;

  for (int t = 0; t < kT; ++t) {
    const size_t bt = (size_t)b * kT + t;
    const float* __restrict__ Qb = Q  + bt * (kN * kN);
    const float* __restrict__ pb = p  + bt * kN;
    const float* __restrict__ Ab = A  + bt * (kNS * kNS);
    const float* __restrict__ Bb = Bm + bt * (kNS * kNC);

    if (t + 1 < kT)
      __builtin_prefetch((const char*)(Qb + kN * kN) + lane * 64, 0, 1);

    // xu[n] = x[n] (n<12) | u[n-12] = cu + k (12<=n<16) | 0
    float xu;
    if (lane < kNS)      xu = x;
    else if (lane < kN)  xu = cu[bt * kNC + (lane - kNS)] +
                              kin[bt * kNC + (lane - kNS)];
    else                 xu = 0.0f;

    if (lane < kNS)                 xs[bt * kNS + lane] = x;
    if (lane >= kNS && lane < kN)   us[bt * kNC + (lane - kNS)] = xu;

    // z[n] = sum_m xu[m] * Q[m][n]  (coalesced row loads per m)
    float z = 0.0f;
    #pragma unroll
    for (int m = 0; m < kN; ++m) {
      const float xm = __shfl(xu, m, 32);
      z += xm * Qb[m * kN + ln];
    }
    cacc += xu * (0.5f * z + pb[ln]);   // xu==0 on lanes >= 16

    // x' = A x + B u
    const int r = (lane < kNS) ? lane : 0;
    float xn = 0.0f;
    #pragma unroll
    for (int s = 0; s < kNS; ++s) {
      const float xv = __shfl(xu, s, 32);
      xn += Ab[r * kNS + s] * xv;
    }
    #pragma unroll
    for (int j = 0; j < kNC; ++j) {
      const float uv = __shfl(xu, kNS + j, 32);
      xn += Bb[r * kNC + j] * uv;
    }
    x = (lane < kNS) ? xn : 0.0f;
  }

  #pragma unroll
  for (int off = 16; off > 0; off >>= 1) cacc += __shfl_xor(cacc, off, 32);
  if (lane == 0) cost[b] = cacc;
}

extern "C" void kernel_launch(void* const* d_in, const int* in_sizes, int n_in,
                              void* d_out, int out_size, void* d_ws, size_t ws_size,
                              hipStream_t stream) {
  (void)in_sizes; (void)n_in; (void)out_size; (void)ws_size;
  const float* x_init = (const float*)d_in[0];
  // d_in[1] = current_x : unused by the reference outputs
  const float* cu     = (const float*)d_in[2];
  const float* Q      = (const float*)d_in[3];
  const float* p      = (const float*)d_in[4];
  const float* A      = (const float*)d_in[5];
  const float* Bm     = (const float*)d_in[6];
  // d_in[7] = time : unused

  float* out  = (float*)d_out;
  float* xs   = out;
  float* us   = out + (size_t)kNB * kT * kNS;
  float* cost = us  + (size_t)kNB * kT * kNC;
  float* kws  = (float*)d_ws;   // NB*T*NC floats = 2 MB of scratch

  dim3 block(256);
  dim3 grid(kNB / WPB);
  lqr_backward<<<grid, block, 0, stream>>>(Q, p, A, Bm, kws);
  lqr_forward <<<grid, block, 0, stream>>>(x_init, cu, Q, p, A, Bm, kws,
                                           xs, us, cost);
}